// TuaAttention_12721693130797
// MI455X (gfx1250) — compile-verified
//
#include <hip/hip_runtime.h>
#include <math.h>

// ---------------------------------------------------------------------------
// Deformable-conv pipeline for MI455X (gfx1250), fp32 WMMA (16x16x4).
// B=4, C=64, H=W=128, K=3, PAD=1.
// Intermediates are NHWC ([b][hw][c]) so deformable gathers and im2col reads
// are contiguous 256B channel runs; offsets are (B,HW,18). Input x is NCHW,
// final output is NCHW (reference layout).
// ---------------------------------------------------------------------------

typedef __attribute__((ext_vector_type(2))) float v2f;
typedef __attribute__((ext_vector_type(8))) float v8f;

#define HWC 16384   // 128*128
#define WID 128
#define NCH 64

__device__ __forceinline__ float gelu_exact(float x) {
    return 0.5f * x * (1.0f + erff(x * 0.70710678118654752f));
}

// ---------------------------------------------------------------------------
// 1x1 conv as WMMA matmul:  out[p,o] = sum_c W[o,c] * (in[p,c] (+in2)) + bias[o]
// block = 256 threads (8 waves). Tile: 64 out-ch x 32 pixels.
// wave w: M-tile = w&3 (16 out ch), N-tile = w>>2 (16 pixels).
// ---------------------------------------------------------------------------
template <int GELU, int IN_NCHW, int OUT_NCHW>
__global__ __launch_bounds__(256) void k_conv1x1(
    const float* __restrict__ in, const float* __restrict__ in2,
    const float* __restrict__ W, const float* __restrict__ bias,
    float* __restrict__ out)
{
    __shared__ float Xs[NCH * 32];          // 8 KB: [c][pixel]

    const long gbase = (long)blockIdx.x * 32;   // pixel index over B*HW
    const int  b  = (int)(gbase / HWC);
    const int  p0 = (int)(gbase % HWC);

    // ---- cooperative load of X tile (64 ch x 32 px), fused residual add ----
    if (IN_NCHW) {
        const int t = threadIdx.x;
        const int c = t >> 2;               // 0..63
        const int q = t & 3;                // 8 pixels each
        const float* src = in + ((size_t)(b * NCH + c) * HWC) + p0 + q * 8;
        float4 v0 = *(const float4*)(src);
        float4 v1 = *(const float4*)(src + 4);
        *(float4*)&Xs[c * 32 + q * 8]     = v0;
        *(float4*)&Xs[c * 32 + q * 8 + 4] = v1;
    } else {
        const int t   = threadIdx.x;
        const int pix = t >> 3;             // 0..31
        const int prt = t & 7;              // 8 channels each
        const float* src = in + ((size_t)(b * HWC + p0 + pix)) * NCH + prt * 8;
        float4 v0 = *(const float4*)(src);
        float4 v1 = *(const float4*)(src + 4);
        if (in2) {
            const float* s2 = in2 + ((size_t)(b * HWC + p0 + pix)) * NCH + prt * 8;
            float4 u0 = *(const float4*)(s2);
            float4 u1 = *(const float4*)(s2 + 4);
            v0.x += u0.x; v0.y += u0.y; v0.z += u0.z; v0.w += u0.w;
            v1.x += u1.x; v1.y += u1.y; v1.z += u1.z; v1.w += u1.w;
        }
        // transpose into [c][pixel]
        Xs[(prt * 8 + 0) * 32 + pix] = v0.x;
        Xs[(prt * 8 + 1) * 32 + pix] = v0.y;
        Xs[(prt * 8 + 2) * 32 + pix] = v0.z;
        Xs[(prt * 8 + 3) * 32 + pix] = v0.w;
        Xs[(prt * 8 + 4) * 32 + pix] = v1.x;
        Xs[(prt * 8 + 5) * 32 + pix] = v1.y;
        Xs[(prt * 8 + 6) * 32 + pix] = v1.z;
        Xs[(prt * 8 + 7) * 32 + pix] = v1.w;
    }
    __syncthreads();

    const int wv    = threadIdx.x >> 5;
    const int lane  = threadIdx.x & 31;
    const int mtile = wv & 3;
    const int ntile = wv >> 2;
    const int n     = lane & 15;            // A: row M, B: column N
    const int hi    = lane >> 4;            // selects K pair
    const int kA    = hi * 2;

    v8f acc;
#pragma unroll
    for (int r = 0; r < 8; ++r)
        acc[r] = bias[mtile * 16 + r + 8 * hi];

    const float* wrow = W + (size_t)(mtile * 16 + n) * NCH;
#pragma unroll
    for (int k0 = 0; k0 < NCH; k0 += 4) {
        v2f a, bm;
        a.x  = wrow[k0 + kA];
        a.y  = wrow[k0 + kA + 1];
        bm.x = Xs[(k0 + kA) * 32 + ntile * 16 + n];
        bm.y = Xs[(k0 + kA + 1) * 32 + ntile * 16 + n];
        acc = __builtin_amdgcn_wmma_f32_16x16x4_f32(
            false, a, false, bm, (short)0, acc, false, false);
    }

    if (OUT_NCHW) {
#pragma unroll
        for (int r = 0; r < 8; ++r) {
            const int och = mtile * 16 + r + 8 * hi;
            float v = acc[r];
            if (GELU) v = gelu_exact(v);
            out[((size_t)(b * NCH + och)) * HWC + p0 + ntile * 16 + n] = v;
        }
    } else {
        float4 f0, f1;
        f0.x = acc[0]; f0.y = acc[1]; f0.z = acc[2]; f0.w = acc[3];
        f1.x = acc[4]; f1.y = acc[5]; f1.z = acc[6]; f1.w = acc[7];
        if (GELU) {
            f0.x = gelu_exact(f0.x); f0.y = gelu_exact(f0.y);
            f0.z = gelu_exact(f0.z); f0.w = gelu_exact(f0.w);
            f1.x = gelu_exact(f1.x); f1.y = gelu_exact(f1.y);
            f1.z = gelu_exact(f1.z); f1.w = gelu_exact(f1.w);
        }
        // 8 consecutive channels per lane in NHWC -> two 32B-aligned b128
        float* dst = out + ((size_t)(b * HWC + p0 + ntile * 16 + n)) * NCH
                         + mtile * 16 + 8 * hi;
        *(float4*)(dst)     = f0;
        *(float4*)(dst + 4) = f1;
    }
}

// ---------------------------------------------------------------------------
// 3x3 conv (pad=1) with 18 output channels via im2col + WMMA.
// Input NHWC, output (B,HW,18). Block = 128 threads (4 waves), 16 pixels.
// Phase 1: build S[576][16] in LDS. Phase 2: waves 0,1 run the K=576 WMMA.
// Wave 0 owns channels 0..15 (all live); wave 1 owns 16..31 (only 16,17
// live: clamped weight loads, statically-masked stores -> no exec juggling).
// ---------------------------------------------------------------------------
__global__ __launch_bounds__(128) void k_conv3x3_off(
    const float* __restrict__ in, const float* __restrict__ W,
    const float* __restrict__ bias, float* __restrict__ out)
{
    __shared__ float S[576 * 16];           // 36 KB: [c*9+kk][pixel]

    const long gp  = (long)blockIdx.x * 16;
    const int  b   = (int)(gp / HWC);
    const int  hw0 = (int)(gp % HWC);
    const int  h   = hw0 >> 7;
    const int  w0  = hw0 & 127;             // multiple of 16; tile on one row

    // ---- phase 1: im2col (144 (pixel,kk) tasks over 128 threads) ----
    for (int task = threadIdx.x; task < 144; task += 128) {
        const int n  = task & 15;
        const int kk = task >> 4;           // 0..8
        const int hh = h + kk / 3 - 1;
        const int ww = w0 + n + kk % 3 - 1;
        float* sp = &S[kk * 16 + n];        // + c*144 per channel
        if (hh >= 0 && hh < 128 && ww >= 0 && ww < 128) {
            const float* src = in + ((size_t)(b * HWC + hh * WID + ww)) * NCH;
#pragma unroll 4
            for (int c4 = 0; c4 < 16; ++c4) {
                float4 v = *(const float4*)(src + c4 * 4);
                sp[(c4 * 4 + 0) * 144] = v.x;
                sp[(c4 * 4 + 1) * 144] = v.y;
                sp[(c4 * 4 + 2) * 144] = v.z;
                sp[(c4 * 4 + 3) * 144] = v.w;
            }
        } else {
            for (int c = 0; c < NCH; ++c) sp[c * 144] = 0.0f;
        }
    }
    __syncthreads();

    // ---- phase 2: WMMA over K=576, 2 M-tiles covering 18 rows ----
    const int wv   = threadIdx.x >> 5;
    const int lane = threadIdx.x & 31;
    if (wv < 2) {
        const int n  = lane & 15;
        const int hi = lane >> 4;
        const int kA = hi * 2;
        const int mrow = min(wv * 16 + n, 17);      // clamp OOB weight rows

        v8f acc;
#pragma unroll
        for (int r = 0; r < 8; ++r)
            acc[r] = bias[min(wv * 16 + r + 8 * hi, 17)];

        const float* wrow = W + (size_t)mrow * 576;
#pragma unroll 4
        for (int k0 = 0; k0 < 576; k0 += 4) {
            v2f a, bm;
            a.x  = wrow[k0 + kA];
            a.y  = wrow[k0 + kA + 1];
            bm.x = S[(k0 + kA) * 16 + n];
            bm.y = S[(k0 + kA + 1) * 16 + n];
            acc = __builtin_amdgcn_wmma_f32_16x16x4_f32(
                false, a, false, bm, (short)0, acc, false, false);
        }

        // (B,HW,18) store. 18*p is even -> float2 stores are 8B-aligned.
        float* dst = out + ((size_t)(b * HWC + hw0 + n)) * 18;
        if (wv == 0) {
            // channels 8*hi + {0..7}: four float2 stores, all live
            float2 s0; s0.x = acc[0]; s0.y = acc[1];
            float2 s1; s1.x = acc[2]; s1.y = acc[3];
            float2 s2; s2.x = acc[4]; s2.y = acc[5];
            float2 s3; s3.x = acc[6]; s3.y = acc[7];
            float* d = dst + 8 * hi;
            *(float2*)(d)     = s0;
            *(float2*)(d + 2) = s1;
            *(float2*)(d + 4) = s2;
            *(float2*)(d + 6) = s3;
        } else if (hi == 0) {
            // wave 1, lanes 0..15: channels 16,17 (r=0,1)
            float2 s0; s0.x = acc[0]; s0.y = acc[1];
            *(float2*)(dst + 16) = s0;
        }
    }
}

// ---------------------------------------------------------------------------
// Deformable 3x3 conv. Input NHWC, offsets (B,HW,18), output NHWC.
// Block = 128 threads (4 waves), 16 pixels. Phase 1: vectorized bilinear
// sampling into S[576][16]. Phase 2: 4 waves x K=576 WMMA (64 out ch).
// ---------------------------------------------------------------------------
__global__ __launch_bounds__(128) void k_deform(
    const float* __restrict__ in, const float* __restrict__ off,
    const float* __restrict__ Wd, const float* __restrict__ bias,
    float* __restrict__ out)
{
    __shared__ float S[576 * 16];           // 36 KB: [c*9+k][pixel]

    const long gp  = (long)blockIdx.x * 16;
    const int  b   = (int)(gp / HWC);
    const int  hw0 = (int)(gp % HWC);
    const int  h   = hw0 >> 7;
    const int  w0  = hw0 & 127;

    // ---- phase 1: bilinear sampling (144 tasks over 128 threads) ----
    const float* xb = in + (size_t)b * HWC * NCH;
    for (int task = threadIdx.x; task < 144; task += 128) {
        const int n = task & 15;
        const int k = task >> 4;            // 0..8
        const int w = w0 + n;

        // (oy,ox) pair is 8B-aligned in the (B,HW,18) layout
        const float2 oyx = *(const float2*)(
            off + ((size_t)(b * HWC + h * WID + w)) * 18 + 2 * k);
        const float py = oyx.x + (float)h + (float)(k / 3) - 1.0f;
        const float px = oyx.y + (float)w + (float)(k % 3) - 1.0f;

        const float y0f = floorf(py), x0f = floorf(px);
        const float wy = py - y0f, wx = px - x0f;
        const int iy0 = (int)y0f, ix0 = (int)x0f;
        const int iy1 = iy0 + 1,  ix1 = ix0 + 1;

        float w00 = (1.0f - wy) * (1.0f - wx);
        float w01 = (1.0f - wy) * wx;
        float w10 = wy * (1.0f - wx);
        float w11 = wy * wx;

        const bool vy0 = (iy0 >= 0) && (iy0 < 128);
        const bool vy1 = (iy1 >= 0) && (iy1 < 128);
        const bool vx0 = (ix0 >= 0) && (ix0 < 128);
        const bool vx1 = (ix1 >= 0) && (ix1 < 128);
        if (!(vy0 && vx0)) w00 = 0.0f;
        if (!(vy0 && vx1)) w01 = 0.0f;
        if (!(vy1 && vx0)) w10 = 0.0f;
        if (!(vy1 && vx1)) w11 = 0.0f;

        const int cy0 = min(max(iy0, 0), 127), cy1 = min(max(iy1, 0), 127);
        const int cx0 = min(max(ix0, 0), 127), cx1 = min(max(ix1, 0), 127);
        const float* p00 = xb + (size_t)(cy0 * WID + cx0) * NCH;
        const float* p01 = xb + (size_t)(cy0 * WID + cx1) * NCH;
        const float* p10 = xb + (size_t)(cy1 * WID + cx0) * NCH;
        const float* p11 = xb + (size_t)(cy1 * WID + cx1) * NCH;

        float* sp = &S[k * 16 + n];         // + c*144 per channel
#pragma unroll 4
        for (int c4 = 0; c4 < 16; ++c4) {
            float4 a0 = *(const float4*)(p00 + c4 * 4);
            float4 a1 = *(const float4*)(p01 + c4 * 4);
            float4 a2 = *(const float4*)(p10 + c4 * 4);
            float4 a3 = *(const float4*)(p11 + c4 * 4);
            sp[(c4 * 4 + 0) * 144] = w00 * a0.x + w01 * a1.x + w10 * a2.x + w11 * a3.x;
            sp[(c4 * 4 + 1) * 144] = w00 * a0.y + w01 * a1.y + w10 * a2.y + w11 * a3.y;
            sp[(c4 * 4 + 2) * 144] = w00 * a0.z + w01 * a1.z + w10 * a2.z + w11 * a3.z;
            sp[(c4 * 4 + 3) * 144] = w00 * a0.w + w01 * a1.w + w10 * a2.w + w11 * a3.w;
        }
    }
    __syncthreads();

    // ---- phase 2: WMMA over K=576 ----
    const int wv   = threadIdx.x >> 5;      // M-tile (16 out ch each)
    const int lane = threadIdx.x & 31;
    const int n    = lane & 15;
    const int hi   = lane >> 4;
    const int kA   = hi * 2;

    v8f acc;
#pragma unroll
    for (int r = 0; r < 8; ++r)
        acc[r] = bias[wv * 16 + r + 8 * hi];

    const float* wrow = Wd + (size_t)(wv * 16 + n) * 576;
    __builtin_prefetch(wrow, 0, 0);                     // global_prefetch_b8
    __builtin_prefetch(wrow + 288, 0, 0);

#pragma unroll 4
    for (int k0 = 0; k0 < 576; k0 += 4) {
        v2f a, bm;
        a.x  = wrow[k0 + kA];
        a.y  = wrow[k0 + kA + 1];
        bm.x = S[(k0 + kA) * 16 + n];
        bm.y = S[(k0 + kA + 1) * 16 + n];
        acc = __builtin_amdgcn_wmma_f32_16x16x4_f32(
            false, a, false, bm, (short)0, acc, false, false);
    }

    // NHWC store: 8 consecutive channels per lane -> two 32B-aligned b128
    float4 f0, f1;
    f0.x = acc[0]; f0.y = acc[1]; f0.z = acc[2]; f0.w = acc[3];
    f1.x = acc[4]; f1.y = acc[5]; f1.z = acc[6]; f1.w = acc[7];
    float* dst = out + ((size_t)(b * HWC + hw0 + n)) * NCH + wv * 16 + 8 * hi;
    *(float4*)(dst)     = f0;
    *(float4*)(dst + 4) = f1;
}

// ---------------------------------------------------------------------------
// Host-side orchestration
// ---------------------------------------------------------------------------
extern "C" void kernel_launch(void* const* d_in, const int* in_sizes, int n_in,
                              void* d_out, int out_size, void* d_ws, size_t ws_size,
                              hipStream_t stream) {
    const float* x   = (const float*)d_in[0];
    const float* w1  = (const float*)d_in[1];
    const float* b1  = (const float*)d_in[2];
    const float* wo1 = (const float*)d_in[3];
    const float* bo1 = (const float*)d_in[4];
    const float* wd1 = (const float*)d_in[5];
    const float* bd1 = (const float*)d_in[6];
    const float* wo2 = (const float*)d_in[7];
    const float* bo2 = (const float*)d_in[8];
    const float* wd2 = (const float*)d_in[9];
    const float* bd2 = (const float*)d_in[10];
    const float* w2  = (const float*)d_in[11];
    const float* b2  = (const float*)d_in[12];
    const float* wf  = (const float*)d_in[13];
    const float* bf  = (const float*)d_in[14];
    float* out = (float*)d_out;

    const size_t TEN = (size_t)4 * NCH * HWC;   // 4,194,304 floats
    float* A    = (float*)d_ws;                 // x_act (NHWC, kept for residual)
    float* Bbuf = A + TEN;                      // NHWC
    float* Cbuf = Bbuf + TEN;                   // NHWC
    float* OFF  = Cbuf + TEN;                   // (B,HW,18)
    (void)ws_size; (void)in_sizes; (void)n_in; (void)out_size;

    const int npix = 4 * HWC;                   // 65536

    // 1) x_act = gelu(conv1x1(x, w1) + b1)        NCHW -> NHWC
    k_conv1x1<1, 1, 0><<<npix / 32, 256, 0, stream>>>(x, nullptr, w1, b1, A);
    // 2) off1 = conv3x3(x_act, wo1) + bo1          NHWC -> (B,HW,18)
    k_conv3x3_off<<<npix / 16, 128, 0, stream>>>(A, wo1, bo1, OFF);
    // 3) x_dcn = deform_conv(x_act, off1, wd1, bd1)
    k_deform<<<npix / 16, 128, 0, stream>>>(A, OFF, wd1, bd1, Bbuf);
    // 4) off2 = conv3x3(x_dcn, wo2) + bo2
    k_conv3x3_off<<<npix / 16, 128, 0, stream>>>(Bbuf, wo2, bo2, OFF);
    // 5) x_dcn = deform_conv(x_dcn, off2, wd2, bd2)
    k_deform<<<npix / 16, 128, 0, stream>>>(Bbuf, OFF, wd2, bd2, Cbuf);
    // 6) x_dcn = conv1x1(x_dcn, w2) + b2
    k_conv1x1<0, 0, 0><<<npix / 32, 256, 0, stream>>>(Cbuf, nullptr, w2, b2, Bbuf);
    // 7) out = conv1x1(x_act + x_dcn, wf) + bf     NHWC -> NCHW (reference layout)
    k_conv1x1<0, 0, 1><<<npix / 32, 256, 0, stream>>>(A, Bbuf, wf, bf, out);
}